// MovingAvgFreq_47339129537134
// MI455X (gfx1250) — compile-verified
//
#include <hip/hip_runtime.h>
#include <cstdint>

// ---------------------------------------------------------------------------
// MovingAvgFreq: out[f,c] = K[f] * x[f,c]  (complex64, planar in -> interleaved out)
//
// Bandwidth-bound kernel (AI ~ 0.375 FLOP/B; ~268 MB total @ 23.3 TB/s => ~11.5 us).
// Strategy (MI455X / gfx1250):
//   * 4 columns per lane -> all global traffic is B128.
//   * CDNA5 async path: GLOBAL_LOAD_ASYNC_TO_LDS_B128 double-buffer,
//     synchronized with S_WAIT_ASYNCCNT (no barriers: each lane reads back
//     exactly the LDS slot it filled, so per-wave counter ordering suffices).
//   * Non-temporal B128 stores (stream-once output, keep 192MB L2 clean).
//   * K[row] is wave-uniform (blockIdx.y) -> scalar s_load path.
// ---------------------------------------------------------------------------

typedef float f32x4 __attribute__((ext_vector_type(4)));
typedef int   v4i   __attribute__((vector_size(4 * sizeof(int))));

#define THREADS          256
#define TILE_COLS        (THREADS * 4)   // 1024 columns per tile
#define TILES_PER_BLOCK  4

#if defined(__gfx1250__) && __has_builtin(__builtin_amdgcn_global_load_async_to_lds_b128)
#define USE_ASYNC_LDS 1
#endif

#ifdef USE_ASYNC_LDS
typedef __attribute__((address_space(1))) v4i* gv4i_ptr;   // global int4*
typedef __attribute__((address_space(3))) v4i* lv4i_ptr;   // LDS int4*
#endif

__device__ __forceinline__ void async_copy16(const float* g, float* lds) {
#ifdef USE_ASYNC_LDS
  // (global src, lds dst, imm offset, imm cpol)
  __builtin_amdgcn_global_load_async_to_lds_b128(
      (gv4i_ptr)(uintptr_t)g,
      (lv4i_ptr)(uint32_t)(uintptr_t)lds,
      0, 0);
#else
  *(f32x4*)lds = *(const f32x4*)g;   // fallback: global load + ds store
#endif
}

template <int N>
__device__ __forceinline__ void wait_async_le() {
#ifdef USE_ASYNC_LDS
  if constexpr (N == 0) {
    asm volatile("s_wait_asynccnt 0x0" ::: "memory");
  } else {
    asm volatile("s_wait_asynccnt 0x2" ::: "memory");
  }
#else
  asm volatile("" ::: "memory");
#endif
}

__global__ __launch_bounds__(THREADS)
void movavg_freq_scale_kernel(const float* __restrict__ xr,
                              const float* __restrict__ xi,
                              const float* __restrict__ Kr,
                              const float* __restrict__ Ki,
                              float* __restrict__ out,
                              int cols, int tiles_per_row)
{
  __shared__ __align__(16) float s_re[2][TILE_COLS];
  __shared__ __align__(16) float s_im[2][TILE_COLS];

  const int row = blockIdx.y;
  const int tid = threadIdx.x;
  const int t0  = blockIdx.x * TILES_PER_BLOCK;

  int ntiles = tiles_per_row - t0;
  if (ntiles <= 0) return;
  if (ntiles > TILES_PER_BLOCK) ntiles = TILES_PER_BLOCK;

  // K[row] is uniform across the block -> scalar loads.
  const float kr = Kr[row];
  const float ki = Ki[row];
  const size_t rowbase = (size_t)row * (size_t)cols;

  // Column for (tile, lane); clamped so the async issue count stays uniform
  // even on a ragged tail (duplicate loads/stores of identical data are benign).
  auto col_of = [&](int tile) -> int {
    long long c = (long long)(t0 + tile) * TILE_COLS + 4 * tid;
    long long cmax = (long long)cols - 4;
    return (int)(c > cmax ? cmax : c);
  };

  // Prologue: stage tile 0 into buffer 0.  (ASYNCcnt = 2)
  {
    const int c = col_of(0);
    async_copy16(xr + rowbase + c, &s_re[0][4 * tid]);
    async_copy16(xi + rowbase + c, &s_im[0][4 * tid]);
  }

  for (int i = 0; i < ntiles; ++i) {
    const int cur = i & 1;

    if (i + 1 < ntiles) {
      // Stage tile i+1 into the other buffer, then wait for tile i only.
      const int c = col_of(i + 1);
      async_copy16(xr + rowbase + c, &s_re[cur ^ 1][4 * tid]);
      async_copy16(xi + rowbase + c, &s_im[cur ^ 1][4 * tid]);
      wait_async_le<2>();          // tile i's 2 ops have completed
    } else {
      wait_async_le<0>();          // drain everything for the final tile
    }

    // Each lane reads back exactly the 16B slots it filled -> no barrier.
    const f32x4 re = *(const f32x4*)&s_re[cur][4 * tid];
    const f32x4 im = *(const f32x4*)&s_im[cur][4 * tid];

    // (kr + i*ki) * (re + i*im)
    const f32x4 orr = kr * re - ki * im;
    const f32x4 oii = kr * im + ki * re;

    // Interleave to complex64 layout: (r0,i0,r1,i1)(r2,i2,r3,i3)
    const f32x4 lo = {orr.x, oii.x, orr.y, oii.y};
    const f32x4 hi = {orr.z, oii.z, orr.w, oii.w};

    const int c = col_of(i);
    float* op = out + 2 * (rowbase + (size_t)c);
    __builtin_nontemporal_store(lo, (f32x4*)op);        // global_store_b128 ... NT
    __builtin_nontemporal_store(hi, (f32x4*)(op + 4));
  }
}

extern "C" void kernel_launch(void* const* d_in, const int* in_sizes, int n_in,
                              void* d_out, int out_size, void* d_ws, size_t ws_size,
                              hipStream_t stream) {
  (void)n_in; (void)out_size; (void)d_ws; (void)ws_size;

  const float* xr = (const float*)d_in[0];   // [F, COLS]
  const float* xi = (const float*)d_in[1];   // [F, COLS]
  const float* Kr = (const float*)d_in[2];   // [F]
  const float* Ki = (const float*)d_in[3];   // [F]
  float* out = (float*)d_out;                // [F, COLS] complex64 -> interleaved f32

  const int F    = in_sizes[2];              // 2049
  const int cols = in_sizes[0] / F;          // 8192

  const int tiles_per_row = (cols + TILE_COLS - 1) / TILE_COLS;           // 8
  const int gx = (tiles_per_row + TILES_PER_BLOCK - 1) / TILES_PER_BLOCK; // 2

  dim3 grid(gx, F, 1);
  dim3 block(THREADS, 1, 1);
  hipLaunchKernelGGL(movavg_freq_scale_kernel, grid, block, 0, stream,
                     xr, xi, Kr, Ki, out, cols, tiles_per_row);
}